// GAT_39977555591268
// MI455X (gfx1250) — compile-verified
//
#include <hip/hip_runtime.h>
#include <math.h>

// ---------------------------------------------------------------------------
// 3-layer GAT forward for MI455X (gfx1250, wave32).
//   GEMMs   : V_WMMA_F32_16X16X4_F32 (exact fp32 tensor path; GEMMs are only
//             ~42 GFLOP so fp32 WMMA is plenty and matches reference precision)
//   Edge ops: L2-resident (h = 102 MB < 192 MB L2) gathers + f32 atomics.
// ---------------------------------------------------------------------------

#define N_NODES   50000
#define E0        800000
#define ETOT      850000            // E0 + N_NODES self loops
#define F_IN      256
#define HID       64
#define HEADS     8
#define HC        512               // HEADS*HID
#define OUTC      47
#define NEG_SLOPE 0.2f

typedef __attribute__((ext_vector_type(2))) float v2f;
typedef __attribute__((ext_vector_type(8))) float v8f;

__device__ __forceinline__ void edge_nodes(int e, const int* __restrict__ ei,
                                           int& s, int& d) {
  if (e < E0) { s = ei[e]; d = ei[E0 + e]; }
  else        { s = e - E0; d = s; }          // appended self loops
}

// float atomic-max via monotonic int/uint bit trick (init with -inf).
__device__ __forceinline__ void atomic_max_float(float* addr, float val) {
  if (val >= 0.0f) atomicMax((int*)addr, __float_as_int(val));
  else             atomicMin((unsigned int*)addr, __float_as_uint(val));
}

// ---------------------------------------------------------------------------
// D[N,M] = A[N,K] @ B[K,M], fp32 WMMA. One wave per 16x16 D tile.
// A-frag: lane<16 -> K={k,k+1}, lane>=16 -> K={k+2,k+3}, row = lane&15.
// B-frag: mirrored (col = lane&15). D: VGPR v -> row v (+8 for upper half).
// BOUNDED=false (layers 1-2, M%16==0): branch-free inner loop so codegen is
// pure load_b64/load_b32 + v_wmma with no exec-mask juggling.
// ---------------------------------------------------------------------------
template <bool BOUNDED>
__global__ void gemm_wmma_f32(const float* __restrict__ A, const float* __restrict__ B,
                              float* __restrict__ D, int K, int M) {
  const int wave = threadIdx.x >> 5;
  const int lane = threadIdx.x & 31;
  const int rt   = blockIdx.x;                 // row tile (N/16 exact: 3125)
  const int ct   = blockIdx.y * 8 + wave;      // col tile
  if (ct * 16 >= M) return;                    // wave-uniform exit
  const int  half = lane >> 4;
  const int  lp   = lane & 15;
  const int  arow = rt * 16 + lp;
  const int  bcol = ct * 16 + lp;
  const float* __restrict__ Ar = A + (size_t)arow * K;

  v8f acc = {};
  if (!BOUNDED) {
    const float* __restrict__ Bc = B + bcol;   // column pointer, stride M
#pragma unroll 4
    for (int k = 0; k < K; k += 4) {
      const int kk = k + half * 2;
      v2f a, b;
      a[0] = Ar[kk];
      a[1] = Ar[kk + 1];
      b[0] = Bc[(size_t)kk * M];
      b[1] = Bc[(size_t)(kk + 1) * M];
      acc = __builtin_amdgcn_wmma_f32_16x16x4_f32(false, a, false, b,
                                                  (short)0, acc, false, false);
    }
  } else {
    const bool bok = bcol < M;                 // M=47 tail guard (layer 3)
    for (int k = 0; k < K; k += 4) {
      const int kk = k + half * 2;
      v2f a, b;
      a[0] = Ar[kk];
      a[1] = Ar[kk + 1];
      b[0] = bok ? B[(size_t)kk * M + bcol]       : 0.0f;
      b[1] = bok ? B[(size_t)(kk + 1) * M + bcol] : 0.0f;
      acc = __builtin_amdgcn_wmma_f32_16x16x4_f32(false, a, false, b,
                                                  (short)0, acc, false, false);
    }
  }
#pragma unroll
  for (int v = 0; v < 8; ++v) {
    const int row = rt * 16 + half * 8 + v;
    if (!BOUNDED || bcol < M) D[(size_t)row * M + bcol] = acc[v];
  }
}

// al_s[n,h] = sum_c h[n,h,c]*a_src[h,c] ; al_d likewise.
__global__ void attn_coef(const float* __restrict__ h, const float* __restrict__ a_s,
                          const float* __restrict__ a_d, float* __restrict__ als,
                          float* __restrict__ ald, int H, int C) {
  const int i = blockIdx.x * blockDim.x + threadIdx.x;   // i = n*H + head
  if (i >= N_NODES * H) return;
  const int head = i % H;
  const int n    = i / H;
  const float* hv  = h   + (size_t)n * H * C + (size_t)head * C;
  const float* asp = a_s + (size_t)head * C;
  const float* adp = a_d + (size_t)head * C;
  float ss = 0.0f, sd = 0.0f;
  for (int c = 0; c < C; ++c) { const float x = hv[c]; ss += x * asp[c]; sd += x * adp[c]; }
  als[i] = ss;
  ald[i] = sd;
}

__global__ void init_md(float* __restrict__ m, float* __restrict__ d, int n) {
  const int i = blockIdx.x * blockDim.x + threadIdx.x;
  if (i < n) { m[i] = -INFINITY; d[i] = 0.0f; }
}

// pass A: e = leaky_relu(al_s[src]+al_d[dst]); ebuf=e; m[dst] = max(m[dst], e)
__global__ void edge_logits(const int* __restrict__ ei, const float* __restrict__ als,
                            const float* __restrict__ ald, float* __restrict__ ebuf,
                            float* __restrict__ m, int H) {
  const int i = blockIdx.x * blockDim.x + threadIdx.x;   // i = e*H + head
  if (i >= ETOT * H) return;
  const int head = i % H;
  const int e    = i / H;
  int s, d; edge_nodes(e, ei, s, d);
  float v = als[s * H + head] + ald[d * H + head];
  v = (v > 0.0f) ? v : NEG_SLOPE * v;
  ebuf[i] = v;
  atomic_max_float(&m[d * H + head], v);
}

// pass B: ex = exp(e - m[dst]); ebuf=ex; denom[dst] += ex
__global__ void edge_exp(const int* __restrict__ ei, float* __restrict__ ebuf,
                         const float* __restrict__ m, float* __restrict__ denom, int H) {
  const int i = blockIdx.x * blockDim.x + threadIdx.x;
  if (i >= ETOT * H) return;
  const int head = i % H;
  const int e    = i / H;
  int s, d; edge_nodes(e, ei, s, d);
  const float ex = __expf(ebuf[i] - m[d * H + head]);
  ebuf[i] = ex;
  atomicAdd(&denom[d * H + head], ex);
}

// pass C (layers 1-2, C=64): acc[dst,h,c] += ex * h[src,h,c], float4 gathers.
__global__ void edge_scatter_v4(const int* __restrict__ ei, const float* __restrict__ ebuf,
                                const float* __restrict__ h, float* __restrict__ acc) {
  const unsigned i = blockIdx.x * blockDim.x + threadIdx.x;  // ETOT*HEADS*(HID/4)
  if (i >= (unsigned)ETOT * HEADS * (HID / 4)) return;
  const int c4   = i & 15;          // HID/4 = 16
  const int head = (i >> 4) & 7;    // HEADS = 8
  const int e    = i >> 7;
  int s, d; edge_nodes(e, ei, s, d);
  const float  ex = ebuf[e * HEADS + head];
  const float4 hv = *(const float4*)(h + (size_t)s * HC + head * HID + c4 * 4);
  float* ap = acc + (size_t)d * HC + head * HID + c4 * 4;
  atomicAdd(ap + 0, ex * hv.x);
  atomicAdd(ap + 1, ex * hv.y);
  atomicAdd(ap + 2, ex * hv.z);
  atomicAdd(ap + 3, ex * hv.w);
}

// pass C (layer 3, H=1, C=47): scalar channels.
__global__ void edge_scatter_s(const int* __restrict__ ei, const float* __restrict__ ebuf,
                               const float* __restrict__ h, float* __restrict__ acc, int C) {
  const unsigned i = blockIdx.x * blockDim.x + threadIdx.x;  // ETOT*C
  if (i >= (unsigned)ETOT * C) return;
  const int c = i % C;
  const int e = i / C;
  int s, d; edge_nodes(e, ei, s, d);
  atomicAdd(&acc[(size_t)d * C + c], ebuf[e] * h[(size_t)s * C + c]);
}

// out = acc/denom + bias (optional relu), in place.
__global__ void finalize(float* __restrict__ acc, const float* __restrict__ denom,
                         const float* __restrict__ bias, int H, int C, int relu) {
  const unsigned i = blockIdx.x * blockDim.x + threadIdx.x;  // N*H*C
  if (i >= (unsigned)N_NODES * H * C) return;
  const int j    = i % (H * C);
  const int n    = i / (H * C);
  const int head = j / C;
  float v = acc[i] / (denom[n * H + head] + 1e-16f) + bias[j];
  if (relu) v = fmaxf(v, 0.0f);
  acc[i] = v;
}

__global__ void log_softmax_rows(const float* __restrict__ x, float* __restrict__ out) {
  const int n = blockIdx.x * blockDim.x + threadIdx.x;
  if (n >= N_NODES) return;
  const float* r = x + (size_t)n * OUTC;
  float mx = r[0];
  for (int c = 1; c < OUTC; ++c) mx = fmaxf(mx, r[c]);
  float s = 0.0f;
  for (int c = 0; c < OUTC; ++c) s += __expf(r[c] - mx);
  const float ls = __logf(s);
  float* o = out + (size_t)n * OUTC;
  for (int c = 0; c < OUTC; ++c) o[c] = r[c] - mx - ls;
}

// ---------------------------------------------------------------------------

static inline unsigned cdiv(size_t a, unsigned b) { return (unsigned)((a + b - 1) / b); }

extern "C" void kernel_launch(void* const* d_in, const int* in_sizes, int n_in,
                              void* d_out, int out_size, void* d_ws, size_t ws_size,
                              hipStream_t stream) {
  const float* x   = (const float*)d_in[0];
  const int*   ei  = (const int*)  d_in[1];
  const float* W1  = (const float*)d_in[2];
  const float* as1 = (const float*)d_in[3];
  const float* ad1 = (const float*)d_in[4];
  const float* b1  = (const float*)d_in[5];
  const float* W2  = (const float*)d_in[6];
  const float* as2 = (const float*)d_in[7];
  const float* ad2 = (const float*)d_in[8];
  const float* b2  = (const float*)d_in[9];
  const float* W3  = (const float*)d_in[10];
  const float* as3 = (const float*)d_in[11];
  const float* ad3 = (const float*)d_in[12];
  const float* b3  = (const float*)d_in[13];

  // Workspace layout (floats), ~341 MB total.
  float* hbuf = (float*)d_ws;                         // N*HC  (post-GEMM features)
  float* bufA = hbuf + (size_t)N_NODES * HC;          // N*HC  (layer out / next in)
  float* bufB = bufA + (size_t)N_NODES * HC;          // N*HC
  float* ebuf = bufB + (size_t)N_NODES * HC;          // ETOT*HEADS edge logits/weights
  float* als  = ebuf + (size_t)ETOT * HEADS;          // N*HEADS
  float* ald  = als  + (size_t)N_NODES * HEADS;       // N*HEADS
  float* mbuf = ald  + (size_t)N_NODES * HEADS;       // N*HEADS
  float* dbuf = mbuf + (size_t)N_NODES * HEADS;       // N*HEADS
  (void)in_sizes; (void)n_in; (void)out_size; (void)ws_size;

  const unsigned T = 256;

  // ---------------- Layer 1: F_IN -> 8 x 64, relu ----------------
  {
    dim3 g(N_NODES / 16, (HC + 127) / 128);
    gemm_wmma_f32<false><<<g, T, 0, stream>>>(x, W1, hbuf, F_IN, HC);
    attn_coef<<<cdiv((size_t)N_NODES * HEADS, T), T, 0, stream>>>(hbuf, as1, ad1, als, ald, HEADS, HID);
    init_md<<<cdiv((size_t)N_NODES * HEADS, T), T, 0, stream>>>(mbuf, dbuf, N_NODES * HEADS);
    hipMemsetAsync(bufA, 0, (size_t)N_NODES * HC * sizeof(float), stream);
    edge_logits<<<cdiv((size_t)ETOT * HEADS, T), T, 0, stream>>>(ei, als, ald, ebuf, mbuf, HEADS);
    edge_exp<<<cdiv((size_t)ETOT * HEADS, T), T, 0, stream>>>(ei, ebuf, mbuf, dbuf, HEADS);
    edge_scatter_v4<<<cdiv((size_t)ETOT * HEADS * (HID / 4), T), T, 0, stream>>>(ei, ebuf, hbuf, bufA);
    finalize<<<cdiv((size_t)N_NODES * HC, T), T, 0, stream>>>(bufA, dbuf, b1, HEADS, HID, 1);
  }

  // ---------------- Layer 2: 512 -> 8 x 64, relu ----------------
  {
    dim3 g(N_NODES / 16, (HC + 127) / 128);
    gemm_wmma_f32<false><<<g, T, 0, stream>>>(bufA, W2, hbuf, HC, HC);
    attn_coef<<<cdiv((size_t)N_NODES * HEADS, T), T, 0, stream>>>(hbuf, as2, ad2, als, ald, HEADS, HID);
    init_md<<<cdiv((size_t)N_NODES * HEADS, T), T, 0, stream>>>(mbuf, dbuf, N_NODES * HEADS);
    hipMemsetAsync(bufB, 0, (size_t)N_NODES * HC * sizeof(float), stream);
    edge_logits<<<cdiv((size_t)ETOT * HEADS, T), T, 0, stream>>>(ei, als, ald, ebuf, mbuf, HEADS);
    edge_exp<<<cdiv((size_t)ETOT * HEADS, T), T, 0, stream>>>(ei, ebuf, mbuf, dbuf, HEADS);
    edge_scatter_v4<<<cdiv((size_t)ETOT * HEADS * (HID / 4), T), T, 0, stream>>>(ei, ebuf, hbuf, bufB);
    finalize<<<cdiv((size_t)N_NODES * HC, T), T, 0, stream>>>(bufB, dbuf, b2, HEADS, HID, 1);
  }

  // ---------------- Layer 3: 512 -> 1 x 47, log_softmax ----------------
  {
    dim3 g(N_NODES / 16, 1);
    gemm_wmma_f32<true><<<g, T, 0, stream>>>(bufB, W3, hbuf, HC, OUTC);
    attn_coef<<<cdiv((size_t)N_NODES, T), T, 0, stream>>>(hbuf, as3, ad3, als, ald, 1, OUTC);
    init_md<<<cdiv((size_t)N_NODES, T), T, 0, stream>>>(mbuf, dbuf, N_NODES);
    hipMemsetAsync(bufA, 0, (size_t)N_NODES * OUTC * sizeof(float), stream);
    edge_logits<<<cdiv((size_t)ETOT, T), T, 0, stream>>>(ei, als, ald, ebuf, mbuf, 1);
    edge_exp<<<cdiv((size_t)ETOT, T), T, 0, stream>>>(ei, ebuf, mbuf, dbuf, 1);
    edge_scatter_s<<<cdiv((size_t)ETOT * OUTC, T), T, 0, stream>>>(ei, ebuf, hbuf, bufA, OUTC);
    finalize<<<cdiv((size_t)N_NODES * OUTC, T), T, 0, stream>>>(bufA, dbuf, b3, 1, OUTC, 0);
    log_softmax_rows<<<cdiv((size_t)N_NODES, T), T, 0, stream>>>(bufA, (float*)d_out);
  }
}